// Block_46273977647542
// MI455X (gfx1250) — compile-verified
//
#include <hip/hip_runtime.h>

// ---------------------------------------------------------------------------
// Sizes from the reference
// ---------------------------------------------------------------------------
constexpr int B_   = 8;
constexpr int N_   = 1024;
constexpr int DIM_ = 768;
constexpr int H_   = 12;
constexpr int CR_  = 384;
constexpr int HD_  = 32;
constexpr float EPS_ = 1e-5f;

typedef __attribute__((ext_vector_type(2))) float v2f;
typedef __attribute__((ext_vector_type(8))) float v8f;

// D = A(16x4) * B(4x16) + C, exact fp32 on the matrix pipe.
static __device__ __forceinline__ v8f wmma4(v2f a, v2f b, v8f c) {
    return __builtin_amdgcn_wmma_f32_16x16x4_f32(
        /*neg_a=*/false, a, /*neg_b=*/false, b,
        /*c_mod=*/(short)0, c, /*reuse_a=*/false, /*reuse_b=*/false);
}

// ---------------------------------------------------------------------------
// LayerNorm over rows of 768 (one 256-thread block per row, 3 elems/thread)
// ---------------------------------------------------------------------------
__global__ __launch_bounds__(256)
void ln768_kernel(const float* __restrict__ x, const float* __restrict__ g,
                  const float* __restrict__ bia, float* __restrict__ o) {
    const int row  = blockIdx.x;
    const int t    = threadIdx.x;
    const float* xr = x + (size_t)row * DIM_;
    float v0 = xr[t], v1 = xr[t + 256], v2 = xr[t + 512];
    float s  = v0 + v1 + v2;
    float s2 = v0 * v0 + v1 * v1 + v2 * v2;
    #pragma unroll
    for (int off = 16; off > 0; off >>= 1) {
        s  += __shfl_xor(s,  off, 32);
        s2 += __shfl_xor(s2, off, 32);
    }
    __shared__ float ps[8], ps2[8];
    const int wid = t >> 5, lane = t & 31;
    if (lane == 0) { ps[wid] = s; ps2[wid] = s2; }
    __syncthreads();
    if (wid == 0) {
        float a  = (lane < 8) ? ps[lane]  : 0.f;
        float a2 = (lane < 8) ? ps2[lane] : 0.f;
        #pragma unroll
        for (int off = 4; off > 0; off >>= 1) {
            a  += __shfl_xor(a,  off, 32);
            a2 += __shfl_xor(a2, off, 32);
        }
        if (lane == 0) { ps[0] = a; ps2[0] = a2; }
    }
    __syncthreads();
    const float mean = ps[0] * (1.f / DIM_);
    const float var  = ps2[0] * (1.f / DIM_) - mean * mean;
    const float rstd = rsqrtf(var + EPS_);
    float* orow = o + (size_t)row * DIM_;
    orow[t]       = (v0 - mean) * rstd * g[t]       + bia[t];
    orow[t + 256] = (v1 - mean) * rstd * g[t + 256] + bia[t + 256];
    orow[t + 512] = (v2 - mean) * rstd * g[t + 512] + bia[t + 512];
}

// ---------------------------------------------------------------------------
// LayerNorm over rows of 384 (one wave per row, 12 elems/lane)
// ---------------------------------------------------------------------------
__global__ __launch_bounds__(256)
void ln384_kernel(const float* __restrict__ x, const float* __restrict__ g,
                  const float* __restrict__ bia, float* __restrict__ o) {
    const int wid  = threadIdx.x >> 5;
    const int lane = threadIdx.x & 31;
    const int row  = blockIdx.x * 8 + wid;
    const float* xr = x + (size_t)row * CR_;
    float v[12];
    float s = 0.f, s2 = 0.f;
    #pragma unroll
    for (int i = 0; i < 12; ++i) {
        v[i] = xr[lane + 32 * i];
        s += v[i]; s2 += v[i] * v[i];
    }
    #pragma unroll
    for (int off = 16; off > 0; off >>= 1) {
        s  += __shfl_xor(s,  off, 32);
        s2 += __shfl_xor(s2, off, 32);
    }
    const float mean = s * (1.f / CR_);
    const float var  = s2 * (1.f / CR_) - mean * mean;
    const float rstd = rsqrtf(var + EPS_);
    float* orow = o + (size_t)row * CR_;
    #pragma unroll
    for (int i = 0; i < 12; ++i) {
        const int c = lane + 32 * i;
        orow[c] = (v[i] - mean) * rstd * g[c] + bia[c];
    }
}

// ---------------------------------------------------------------------------
// GEMM: C[M,N] = A[M,K] * W[N,K]^T (+bias).
// 8 waves/WG, WG tile 128(M) x 64(N); wave = 16 rows x 64 cols.
// K processed in chunks of 32; the 64x32 weight chunk is cooperatively
// staged in double-buffered LDS (row stride 34 -> conflict-free b64 frags),
// next chunk's global loads are issued before the current chunk's 32 WMMAs.
// ---------------------------------------------------------------------------
constexpr int KC      = 32;        // K chunk
constexpr int WSTRIDE = KC + 2;    // 34: 16-lane strided b64 reads hit 16 banks

__global__ __launch_bounds__(256)
void gemm_wmma_kernel(const float* __restrict__ A, const float* __restrict__ W,
                      const float* __restrict__ bias, float* __restrict__ C,
                      int M, int N, int K) {
    __shared__ float sW[2][64 * WSTRIDE];
    const int t    = threadIdx.x;
    const int lane = t & 31;
    const int w    = t >> 5;             // 0..7
    const int hi   = lane >> 4;
    const int lo   = lane & 15;
    const int m0   = blockIdx.y * 128 + w * 16;
    const int n0   = blockIdx.x * 64;
    const int NC   = K / KC;

    const float* arow = A + (size_t)(m0 + lo) * K + 2 * hi;

    float wreg[8], wregN[8];
    v2f   areg[8], aregN[8];

    auto gload = [&](int kc, float* wr, v2f* ar) {
        #pragma unroll
        for (int i = 0; i < 8; ++i)      // W chunk: 8 floats/thread, coalesced
            wr[i] = W[(size_t)(n0 + i * 8 + w) * K + kc + lane];
        #pragma unroll
        for (int ts = 0; ts < 8; ++ts) { // A chunk: 8 b64 frags/lane
            ar[ts].x = arow[kc + 4 * ts];
            ar[ts].y = arow[kc + 4 * ts + 1];
        }
    };

    v8f acc[4] = {};
    gload(0, wreg, areg);

    for (int c = 0; c < NC; ++c) {
        const int bufi = c & 1;
        if (c + 1 < NC) gload((c + 1) * KC, wregN, aregN);  // overlap w/ WMMAs
        #pragma unroll
        for (int i = 0; i < 8; ++i)
            sW[bufi][(i * 8 + w) * WSTRIDE + lane] = wreg[i];
        __syncthreads();
        const float* sbase = sW[bufi];
        #pragma unroll
        for (int ks = 0; ks < KC / 4; ++ks) {
            #pragma unroll
            for (int tt = 0; tt < 4; ++tt) {
                const float* srow = sbase + (tt * 16 + lo) * WSTRIDE + ks * 4 + 2 * hi;
                v2f b; b.x = srow[0]; b.y = srow[1];
                acc[tt] = wmma4(areg[ks], b, acc[tt]);
            }
        }
        if (c + 1 < NC) {
            #pragma unroll
            for (int i = 0; i < 8; ++i) { wreg[i] = wregN[i]; areg[i] = aregN[i]; }
        }
    }

    #pragma unroll
    for (int tt = 0; tt < 4; ++tt) {
        const int col = n0 + tt * 16 + lo;
        const float bv = bias ? bias[col] : 0.f;
        #pragma unroll
        for (int r = 0; r < 8; ++r) {
            const int row = m0 + r + 8 * hi;
            C[(size_t)row * N + col] = acc[tt][r] + bv;
        }
    }
}

// ---------------------------------------------------------------------------
// Attention: one WG per (b,h, 64-query block); 4 waves, each wave owns a
// 16x1024 score strip held entirely in LDS (CDNA5 320KB WGP LDS).  Exact
// one-pass softmax; P*V re-reads the strip in A-layout (LDS bounce = free
// transpose).  Key/value fragment loads are register double-buffered so
// global latency overlaps the WMMA chains.  out = h + attn.
// ---------------------------------------------------------------------------
constexpr int SROW = 1028;                         // padded row stride (banks)
constexpr int ATTN_SMEM = 4 * 16 * SROW * 4;       // 263168 B

__global__ __launch_bounds__(128)
void attn_kernel(const float* __restrict__ Qb, const float* __restrict__ Kb,
                 const float* __restrict__ Vb, const float* __restrict__ Hb,
                 float* __restrict__ Out) {
    extern __shared__ float sS[];
    const int lane = threadIdx.x & 31;
    const int w    = threadIdx.x >> 5;             // 0..3
    const int bh   = blockIdx.y;                   // 0..95
    const int b    = bh / H_;
    const int h    = bh % H_;
    const int q0   = blockIdx.x * 64 + w * 16;
    const int hi   = lane >> 4;
    const int lo   = lane & 15;
    const float scale = 0.17677669529663687f;      // 1/sqrt(32)
    float* sP = sS + (size_t)w * 16 * SROW;

    // Preload the 8 Q fragments for this wave's 16 rows (K = head dim 32).
    const float* qrow = Qb + ((size_t)(b * N_ + q0 + lo) * CR_) + h * HD_ + 2 * hi;
    v2f aq[8];
    #pragma unroll
    for (int ts = 0; ts < 8; ++ts) { aq[ts].x = qrow[4 * ts]; aq[ts].y = qrow[4 * ts + 1]; }

    // --- S = Q K^T * scale, strip 16 x 1024; K frags double-buffered ------
    const float* kbase = Kb + (size_t)b * N_ * CR_ + h * HD_ + 2 * hi;
    v2f bk[8], bkN[8];
    {
        const float* kr = kbase + (size_t)lo * CR_;
        #pragma unroll
        for (int ts = 0; ts < 8; ++ts) { bk[ts].x = kr[4 * ts]; bk[ts].y = kr[4 * ts + 1]; }
    }
    for (int j = 0; j < N_ / 16; ++j) {
        if (j + 1 < N_ / 16) {
            const float* kr = kbase + (size_t)((j + 1) * 16 + lo) * CR_;
            #pragma unroll
            for (int ts = 0; ts < 8; ++ts) { bkN[ts].x = kr[4 * ts]; bkN[ts].y = kr[4 * ts + 1]; }
        }
        v8f acc = {};
        #pragma unroll
        for (int ts = 0; ts < 8; ++ts) acc = wmma4(aq[ts], bk[ts], acc);
        #pragma unroll
        for (int r = 0; r < 8; ++r)
            sP[(r + 8 * hi) * SROW + j * 16 + lo] = acc[r] * scale;
        if (j + 1 < N_ / 16) {
            #pragma unroll
            for (int ts = 0; ts < 8; ++ts) bk[ts] = bkN[ts];
        }
    }

    // --- exact softmax over each of the 16 rows (whole wave per row) ------
    for (int r = 0; r < 16; ++r) {
        float* srow = sP + r * SROW;
        float vals[32];
        float mx = -3.0e38f;
        #pragma unroll
        for (int i = 0; i < 32; ++i) {
            vals[i] = srow[lane + 32 * i];
            mx = fmaxf(mx, vals[i]);
        }
        #pragma unroll
        for (int off = 16; off > 0; off >>= 1)
            mx = fmaxf(mx, __shfl_xor(mx, off, 32));
        float sum = 0.f;
        #pragma unroll
        for (int i = 0; i < 32; ++i) {
            vals[i] = __expf(vals[i] - mx);
            sum += vals[i];
        }
        #pragma unroll
        for (int off = 16; off > 0; off >>= 1)
            sum += __shfl_xor(sum, off, 32);
        const float inv = 1.f / sum;
        #pragma unroll
        for (int i = 0; i < 32; ++i)
            srow[lane + 32 * i] = vals[i] * inv;
    }

    // --- O = P V (16 x 32): 2 K-steps/iter, P(LDS)+V(global) prefetch -----
    v8f oacc[2] = {};
    const float* vbase = Vb + (size_t)b * N_ * CR_ + h * HD_;
    const float* prow  = sP + lo * SROW + 2 * hi;
    const int d0 = lo, d1 = 16 + lo;

    v2f ap[2], bva[2], bvb[2], apN[2], bvaN[2], bvbN[2];
    {
        #pragma unroll
        for (int u = 0; u < 2; ++u) {
            const int k0 = 4 * u;
            ap[u].x = prow[k0]; ap[u].y = prow[k0 + 1];
            const float* vr = vbase + (size_t)(k0 + 2 * hi) * CR_;
            bva[u].x = vr[d0]; bva[u].y = vr[CR_ + d0];
            bvb[u].x = vr[d1]; bvb[u].y = vr[CR_ + d1];
        }
    }
    for (int k0 = 0; k0 < N_; k0 += 8) {
        if (k0 + 8 < N_) {
            #pragma unroll
            for (int u = 0; u < 2; ++u) {
                const int kn = k0 + 8 + 4 * u;
                apN[u].x = prow[kn]; apN[u].y = prow[kn + 1];
                const float* vr = vbase + (size_t)(kn + 2 * hi) * CR_;
                bvaN[u].x = vr[d0]; bvaN[u].y = vr[CR_ + d0];
                bvbN[u].x = vr[d1]; bvbN[u].y = vr[CR_ + d1];
            }
        }
        #pragma unroll
        for (int u = 0; u < 2; ++u) {
            oacc[0] = wmma4(ap[u], bva[u], oacc[0]);
            oacc[1] = wmma4(ap[u], bvb[u], oacc[1]);
        }
        if (k0 + 8 < N_) {
            #pragma unroll
            for (int u = 0; u < 2; ++u) { ap[u] = apN[u]; bva[u] = bvaN[u]; bvb[u] = bvbN[u]; }
        }
    }

    // --- epilogue: out = h + attn ----------------------------------------
    #pragma unroll
    for (int tt = 0; tt < 2; ++tt) {
        const int col = h * HD_ + tt * 16 + lo;
        #pragma unroll
        for (int r = 0; r < 8; ++r) {
            const int row = q0 + r + 8 * hi;
            const size_t idx = (size_t)(b * N_ + row) * CR_ + col;
            Out[idx] = Hb[idx] + oacc[tt][r];
        }
    }
}

// ---------------------------------------------------------------------------
// Host launcher
// ---------------------------------------------------------------------------
extern "C" void kernel_launch(void* const* d_in, const int* in_sizes, int n_in,
                              void* d_out, int out_size, void* d_ws, size_t ws_size,
                              hipStream_t stream) {
    (void)in_sizes; (void)n_in; (void)out_size; (void)ws_size;
    const float* x      = (const float*)d_in[0];
    const float* ln0_g  = (const float*)d_in[1];
    const float* ln0_b  = (const float*)d_in[2];
    const float* w_proj = (const float*)d_in[3];
    const float* b_proj = (const float*)d_in[4];
    const float* ln1_g  = (const float*)d_in[5];
    const float* ln1_b  = (const float*)d_in[6];
    const float* wq     = (const float*)d_in[7];
    const float* wk     = (const float*)d_in[8];
    const float* wv     = (const float*)d_in[9];
    float* out = (float*)d_out;

    // Workspace layout (floats): xn[8192x768] h[8192x384] y q k v (~88 MB)
    float* ws = (float*)d_ws;
    const size_t rowsBig = (size_t)B_ * N_ * DIM_;   // 6,291,456
    const size_t rowsCr  = (size_t)B_ * N_ * CR_;    // 3,145,728
    float* xn = ws;
    float* hB = xn + rowsBig;
    float* yB = hB + rowsCr;
    float* qB = yB + rowsCr;
    float* kB = qB + rowsCr;
    float* vB = kB + rowsCr;

    const int M = B_ * N_;                            // 8192

    // 1) LN over DIM
    ln768_kernel<<<M, 256, 0, stream>>>(x, ln0_g, ln0_b, xn);

    // 2) h = xn @ w_proj^T + b_proj   (M=8192, N=384, K=768)
    gemm_wmma_kernel<<<dim3(CR_ / 64, M / 128), 256, 0, stream>>>(
        xn, w_proj, b_proj, hB, M, CR_, DIM_);

    // 3) y = LN(h) over CR
    ln384_kernel<<<M / 8, 256, 0, stream>>>(hB, ln1_g, ln1_b, yB);

    // 4) Q/K/V = y @ w^T   (M=8192, N=384, K=384)
    gemm_wmma_kernel<<<dim3(CR_ / 64, M / 128), 256, 0, stream>>>(
        yB, wq, nullptr, qB, M, CR_, CR_);
    gemm_wmma_kernel<<<dim3(CR_ / 64, M / 128), 256, 0, stream>>>(
        yB, wk, nullptr, kB, M, CR_, CR_);
    gemm_wmma_kernel<<<dim3(CR_ / 64, M / 128), 256, 0, stream>>>(
        yB, wv, nullptr, vB, M, CR_, CR_);

    // 5) out = h + attention(y)   — needs 257 KB dynamic LDS (CDNA5 WGP).
    (void)hipFuncSetAttribute((const void*)attn_kernel,
                              hipFuncAttributeMaxDynamicSharedMemorySize,
                              ATTN_SMEM);
    attn_kernel<<<dim3(N_ / 64, B_ * H_), 128, ATTN_SMEM, stream>>>(
        qB, kB, vB, hB, out);
}